// DeepseekV3DecoderLayer_41420664603105
// MI455X (gfx1250) — compile-verified
//
#include <hip/hip_runtime.h>
#include <hip/hip_bf16.h>
#include <math.h>

// ---------------- CDNA5 fp32 WMMA types ----------------
typedef float v2f __attribute__((ext_vector_type(2)));
typedef float v8f __attribute__((ext_vector_type(8)));

#define NH_C     16
#define DQK_C    192
#define DV_C     128
#define KV_LORA_C 512
#define Q_LORA_C  1536
#define N_EXP_C   8
#define INTER_C   768
#define H_C       2048
#define SCALING_C 0.07216878364870322f   // 192^-0.5

// =====================================================================
// Tiled GEMM:  C[M,N] (+)= A[M,K] @ W[N,K]^T   (fp32 WMMA 16x16x4)
// Block 256 threads = 8 waves (4 M x 2 N); wave tile 32x32 (2x2 v8f).
// BM=128, BN=64, BK=32.  Double-buffered LDS software pipeline:
// next tile's global loads are issued before the 32-WMMA block, stored
// to the alternate buffer after it; one barrier per K-step.
// mode: 0 = store, 1 = C += acc, 2 = C += rowscale[row*rsStride] * acc
// =====================================================================
#define GBM 128
#define GBN 64
#define GBK 32

__global__ __launch_bounds__(256)
void gemm_wmma_f32(const float* __restrict__ A, int lda,
                   const float* __restrict__ W, int ldw,
                   float* __restrict__ C, int ldc,
                   int M, int N, int K,
                   int mode, const float* __restrict__ rowscale, int rsStride)
{
    __shared__ float As[2][GBM][GBK + 1];
    __shared__ float Bs[2][GBN][GBK + 1];

    const int tid  = threadIdx.x;
    const int wid  = tid >> 5;      // 0..7
    const int lane = tid & 31;
    const int wm   = wid >> 1;      // 0..3  (M sub-tile of 32)
    const int wn   = wid & 1;       // 0..1  (N sub-tile of 32)
    const int m0   = blockIdx.y * GBM;
    const int n0   = blockIdx.x * GBN;
    const int r    = lane & 15;
    const int hi   = lane >> 4;

    // staging indices (A: 4 float4/thread, B: 2 float4/thread)
    const int arow0 = tid >> 3;           // +32*i
    const int akq   = (tid & 7) << 2;

    v8f acc[2][2];
    #pragma unroll
    for (int i = 0; i < 2; ++i)
        #pragma unroll
        for (int j = 0; j < 2; ++j) acc[i][j] = (v8f){};

    const int nsteps = K / GBK;

    // ---- prologue: stage tile 0 into buffer 0 ----
    {
        #pragma unroll
        for (int i = 0; i < 4; ++i) {
            const int row = arow0 + 32 * i;
            const float4 v = *(const float4*)(A + (size_t)(m0 + row) * lda + akq);
            As[0][row][akq + 0] = v.x; As[0][row][akq + 1] = v.y;
            As[0][row][akq + 2] = v.z; As[0][row][akq + 3] = v.w;
        }
        #pragma unroll
        for (int i = 0; i < 2; ++i) {
            const int row = arow0 + 32 * i;
            const float4 v = *(const float4*)(W + (size_t)(n0 + row) * ldw + akq);
            Bs[0][row][akq + 0] = v.x; Bs[0][row][akq + 1] = v.y;
            Bs[0][row][akq + 2] = v.z; Bs[0][row][akq + 3] = v.w;
        }
    }
    __syncthreads();

    for (int s = 0; s < nsteps; ++s) {
        const int buf = s & 1;
        const bool hasNext = (s + 1) < nsteps;

        // ---- issue next tile's global loads (latency overlaps WMMAs) ----
        float4 ra[4], rb[2];
        if (hasNext) {
            const int k0n = (s + 1) * GBK;
            #pragma unroll
            for (int i = 0; i < 4; ++i) {
                const int row = arow0 + 32 * i;
                ra[i] = *(const float4*)(A + (size_t)(m0 + row) * lda + k0n + akq);
            }
            #pragma unroll
            for (int i = 0; i < 2; ++i) {
                const int row = arow0 + 32 * i;
                rb[i] = *(const float4*)(W + (size_t)(n0 + row) * ldw + k0n + akq);
            }
        }

        // ---- batch-load fragments for this K-step ----
        v2f af[2][8], bf[2][8];
        #pragma unroll
        for (int kk = 0; kk < 8; ++kk) {
            const int ka = (kk << 2) + 2 * hi;
            af[0][kk].x = As[buf][wm * 32 + r][ka];
            af[0][kk].y = As[buf][wm * 32 + r][ka + 1];
            af[1][kk].x = As[buf][wm * 32 + 16 + r][ka];
            af[1][kk].y = As[buf][wm * 32 + 16 + r][ka + 1];
            bf[0][kk].x = Bs[buf][wn * 32 + r][ka];
            bf[0][kk].y = Bs[buf][wn * 32 + r][ka + 1];
            bf[1][kk].x = Bs[buf][wn * 32 + 16 + r][ka];
            bf[1][kk].y = Bs[buf][wn * 32 + 16 + r][ka + 1];
        }
        // ---- 32 WMMAs, 2x2 outer product per k-chunk ----
        #pragma unroll
        for (int kk = 0; kk < 8; ++kk) {
            #pragma unroll
            for (int mi = 0; mi < 2; ++mi)
                #pragma unroll
                for (int ni = 0; ni < 2; ++ni)
                    acc[mi][ni] = __builtin_amdgcn_wmma_f32_16x16x4_f32(
                        false, af[mi][kk], false, bf[ni][kk],
                        (short)0, acc[mi][ni], false, false);
        }

        // ---- stash next tile into the alternate buffer ----
        if (hasNext) {
            const int nb = buf ^ 1;
            #pragma unroll
            for (int i = 0; i < 4; ++i) {
                const int row = arow0 + 32 * i;
                As[nb][row][akq + 0] = ra[i].x; As[nb][row][akq + 1] = ra[i].y;
                As[nb][row][akq + 2] = ra[i].z; As[nb][row][akq + 3] = ra[i].w;
            }
            #pragma unroll
            for (int i = 0; i < 2; ++i) {
                const int row = arow0 + 32 * i;
                Bs[nb][row][akq + 0] = rb[i].x; Bs[nb][row][akq + 1] = rb[i].y;
                Bs[nb][row][akq + 2] = rb[i].z; Bs[nb][row][akq + 3] = rb[i].w;
            }
        }
        __syncthreads();
    }

    // ---- epilogue: D layout = lane r -> col, VGPR i + 8*hi -> row ----
    #pragma unroll
    for (int mi = 0; mi < 2; ++mi) {
        #pragma unroll
        for (int ni = 0; ni < 2; ++ni) {
            const int col = n0 + wn * 32 + ni * 16 + r;
            #pragma unroll
            for (int i = 0; i < 8; ++i) {
                const int row = m0 + wm * 32 + mi * 16 + i + 8 * hi;
                float* cp = C + (size_t)row * ldc + col;
                const float v = acc[mi][ni][i];
                if (mode == 0)      *cp = v;
                else if (mode == 1) *cp += v;
                else                *cp += rowscale[(size_t)row * rsStride] * v;
            }
        }
    }
}

// =====================================================================
// RMS norm over last dim (stride-aware, in-place safe).
// =====================================================================
__global__ __launch_bounds__(256)
void rmsnorm_kernel(const float* __restrict__ x, int xs,
                    float* __restrict__ y, int ys,
                    const float* __restrict__ w, int L)
{
    __shared__ float red[256];
    const float* xr = x + (size_t)blockIdx.x * xs;
    float*       yr = y + (size_t)blockIdx.x * ys;
    float s = 0.f;
    for (int i = threadIdx.x; i < L; i += 256) { float v = xr[i]; s += v * v; }
    red[threadIdx.x] = s; __syncthreads();
    for (int o = 128; o > 0; o >>= 1) {
        if (threadIdx.x < o) red[threadIdx.x] += red[threadIdx.x + o];
        __syncthreads();
    }
    const float inv = rsqrtf(red[0] / (float)L + 1e-6f);
    for (int i = threadIdx.x; i < L; i += 256) yr[i] = xr[i] * inv * w[i];
}

// =====================================================================
// Interleaved RoPE, in place.  q: [T][NH*192] (rot at +128 per head),
// ckv: [T][576] (k_rot at +512).  cos/sin: [T][64] with c[j]==c[j+32].
// =====================================================================
__global__ __launch_bounds__(256)
void rope_kernel(float* __restrict__ q, float* __restrict__ ckv,
                 const float* __restrict__ cb, const float* __restrict__ sb)
{
    __shared__ float stg[NH_C * 64 + 64];
    const int t = blockIdx.x;
    float* qt = q   + (size_t)t * (NH_C * DQK_C);
    float* kr = ckv + (size_t)t * 576 + 512;
    const float* c = cb + (size_t)t * 64;
    const float* s = sb + (size_t)t * 64;

    for (int i = threadIdx.x; i < NH_C * 64; i += 256)
        stg[i] = qt[(i >> 6) * DQK_C + 128 + (i & 63)];
    for (int i = threadIdx.x; i < 64; i += 256)
        stg[NH_C * 64 + i] = kr[i];
    __syncthreads();

    for (int i = threadIdx.x; i < NH_C * 64; i += 256) {
        const int h = i >> 6, d = i & 63, j = d & 31;
        const float cv = c[j], sv = s[j];
        const float e = stg[h * 64 + 2 * j], o = stg[h * 64 + 2 * j + 1];
        qt[h * DQK_C + 128 + d] = (d < 32) ? (e * cv - o * sv) : (o * cv + e * sv);
    }
    for (int i = threadIdx.x; i < 64; i += 256) {
        const int j = i & 31;
        const float cv = c[j], sv = s[j];
        const float e = stg[NH_C * 64 + 2 * j], o = stg[NH_C * 64 + 2 * j + 1];
        kr[i] = (i < 32) ? (e * cv - o * sv) : (o * cv + e * sv);
    }
}

// =====================================================================
// Flash attention: block = 4 waves = 4 consecutive 16-query tiles of one
// (b,h); K/V LDS tiles shared across the 4 waves.  fp32 WMMA for QK^T
// and PV, batched fragment loads, online softmax via shfl in row-halves.
// Next key tile is prefetched (global_prefetch) while this tile computes.
// =====================================================================
__global__ __launch_bounds__(128)
void attn_kernel(const float* __restrict__ q,    // [T][NH*192] (roped)
                 const float* __restrict__ kv,   // [T][NH*256]
                 const float* __restrict__ ckv,  // [T][576] (roped k_rot @512)
                 float* __restrict__ out,        // [T][NH*128]
                 int S)
{
    __shared__ float qs[4][16][DQK_C + 1];   // stride 193
    __shared__ float ks[16][DQK_C + 1];
    __shared__ float vs[16][DV_C + 1];       // stride 129
    __shared__ float ps[4][16][17];

    const int qt0 = blockIdx.x * 4;
    const int h = blockIdx.y, b = blockIdx.z;
    const int tid  = threadIdx.x;
    const int w    = tid >> 5;          // wave id = q-tile within block
    const int lane = tid & 31;
    const int r = lane & 15, hi = lane >> 4;
    const int qt = qt0 + w;

    // ---- stage 4 Q tiles (shared load across the block) ----
    for (int i = tid; i < 4 * 16 * DQK_C; i += 128) {
        const int ww  = i / (16 * DQK_C);
        const int rem = i % (16 * DQK_C);
        const int rr  = rem / DQK_C, d = rem % DQK_C;
        const size_t t = (size_t)b * S + (qt0 + ww) * 16 + rr;
        qs[ww][rr][d] = q[(t * NH_C + h) * DQK_C + d];
    }
    __syncthreads();

    v8f o[8];
    #pragma unroll
    for (int i = 0; i < 8; ++i) o[i] = (v8f){};
    float mrun[8], lrun[8];
    #pragma unroll
    for (int i = 0; i < 8; ++i) { mrun[i] = -1e30f; lrun[i] = 0.f; }

    const int ktMax = qt0 + 3;
    for (int kt = 0; kt <= ktMax; ++kt) {
        // ---- stage K (16x192) and V (16x128) tiles, block-wide ----
        for (int i = tid; i < 16 * DQK_C; i += 128) {
            const int rr = i / DQK_C, d = i % DQK_C;
            const size_t t = (size_t)b * S + kt * 16 + rr;
            ks[rr][d] = (d < 128) ? kv[(t * NH_C + h) * 256 + d]
                                  : ckv[t * 576 + 512 + (d - 128)];
        }
        for (int i = tid; i < 16 * DV_C; i += 128) {
            const int rr = i >> 7, d = i & 127;
            const size_t t = (size_t)b * S + kt * 16 + rr;
            vs[rr][d] = kv[(t * NH_C + h) * 256 + 128 + d];
        }
        // ---- prefetch next key tile (gfx1250 global_prefetch path) ----
        if (kt < ktMax) {
            const int rr = tid >> 3, seg = tid & 7;            // 16 rows x 8 lines
            const size_t t2 = (size_t)b * S + (kt + 1) * 16 + rr;
            __builtin_prefetch(kv + (t2 * NH_C + h) * 256 + seg * 32, 0, 3);
            if (tid < 32) {                                    // k_rot: 16 rows x 2 lines
                const int rr2 = tid >> 1, seg2 = tid & 1;
                const size_t t3 = (size_t)b * S + (kt + 1) * 16 + rr2;
                __builtin_prefetch(ckv + t3 * 576 + 512 + seg2 * 32, 0, 3);
            }
        }
        __syncthreads();

        if (kt <= qt) {   // wave-uniform branch: EXEC stays all-ones
            // ---- scores = Q K^T : 48 k-chunks in 6 batched groups ----
            v8f sc = {};
            #pragma unroll
            for (int g = 0; g < 6; ++g) {
                v2f af[8], bf[8];
                #pragma unroll
                for (int kk = 0; kk < 8; ++kk) {
                    const int ka = g * 32 + (kk << 2) + 2 * hi;
                    af[kk].x = qs[w][r][ka]; af[kk].y = qs[w][r][ka + 1];
                    bf[kk].x = ks[r][ka];    bf[kk].y = ks[r][ka + 1];
                }
                #pragma unroll
                for (int kk = 0; kk < 8; ++kk)
                    sc = __builtin_amdgcn_wmma_f32_16x16x4_f32(
                             false, af[kk], false, bf[kk], (short)0, sc, false, false);
            }

            // ---- scale + causal mask ----
            const int key = kt * 16 + r;
            #pragma unroll
            for (int i = 0; i < 8; ++i) {
                const int qrow = qt * 16 + i + 8 * hi;
                float v = sc[i] * SCALING_C;
                if (key > qrow) v = -1e30f;
                sc[i] = v;
            }

            // ---- online softmax (row lives across 16 lanes of a half) ----
            #pragma unroll
            for (int i = 0; i < 8; ++i) {
                float v = sc[i];
                for (int m = 1; m < 16; m <<= 1)
                    v = fmaxf(v, __shfl_xor(v, m, 32));
                const float mnew = fmaxf(mrun[i], v);
                const float corr = __expf(mrun[i] - mnew);
                const float p = __expf(sc[i] - mnew);
                sc[i] = p;
                float rs = p;
                for (int m = 1; m < 16; m <<= 1) rs += __shfl_xor(rs, m, 32);
                lrun[i] = lrun[i] * corr + rs;
                mrun[i] = mnew;
                #pragma unroll
                for (int nt = 0; nt < 8; ++nt) o[nt][i] *= corr;
            }

            // ---- P tile to per-wave LDS (D -> A operand re-layout) ----
            #pragma unroll
            for (int i = 0; i < 8; ++i) ps[w][i + 8 * hi][r] = sc[i];
            asm volatile("" ::: "memory");   // wave-local DS is in-order

            // ---- O += P V : P fragments loaded once, reused over 8 dv tiles
            v2f pf[4];
            #pragma unroll
            for (int kc = 0; kc < 4; ++kc) {
                const int ka = (kc << 2) + 2 * hi;
                pf[kc].x = ps[w][r][ka]; pf[kc].y = ps[w][r][ka + 1];
            }
            #pragma unroll
            for (int nt = 0; nt < 8; ++nt) {
                v2f vf[4];
                #pragma unroll
                for (int kc = 0; kc < 4; ++kc) {
                    const int ka = (kc << 2) + 2 * hi;
                    vf[kc].x = vs[ka][nt * 16 + r];
                    vf[kc].y = vs[ka + 1][nt * 16 + r];
                }
                #pragma unroll
                for (int kc = 0; kc < 4; ++kc)
                    o[nt] = __builtin_amdgcn_wmma_f32_16x16x4_f32(
                                false, pf[kc], false, vf[kc], (short)0, o[nt], false, false);
            }
        }
        __syncthreads();
    }

    #pragma unroll
    for (int i = 0; i < 8; ++i) {
        const size_t t = (size_t)b * S + qt * 16 + i + 8 * hi;
        const float inv = 1.0f / lrun[i];
        #pragma unroll
        for (int nt = 0; nt < 8; ++nt)
            out[(t * NH_C + h) * DV_C + nt * 16 + r] = o[nt][i] * inv;
    }
}

// =====================================================================
// Router: sigmoid -> +bias -> group top2 (4 groups of 2) -> expert top2
// -> renorm * 2.5; dense per-token weight row (ties -> lowest index).
// =====================================================================
__global__ __launch_bounds__(256)
void router_kernel(const float* __restrict__ hb, const float* __restrict__ rw,
                   const float* __restrict__ rb, float* __restrict__ dw, int H)
{
    __shared__ float red[256];
    __shared__ float logits[N_EXP_C];
    const int t = blockIdx.x;
    const float* x = hb + (size_t)t * H;
    for (int e = 0; e < N_EXP_C; ++e) {
        float s = 0.f;
        const float* w = rw + (size_t)e * H;
        for (int i = threadIdx.x; i < H; i += 256) s += x[i] * w[i];
        red[threadIdx.x] = s; __syncthreads();
        for (int o = 128; o > 0; o >>= 1) {
            if (threadIdx.x < o) red[threadIdx.x] += red[threadIdx.x + o];
            __syncthreads();
        }
        if (threadIdx.x == 0) logits[e] = red[0];
        __syncthreads();
    }
    if (threadIdx.x == 0) {
        float sr[8], sc[8];
        for (int e = 0; e < 8; ++e) {
            sr[e] = 1.f / (1.f + __expf(-logits[e]));
            sc[e] = sr[e] + rb[e];
        }
        float gs[4];
        for (int g = 0; g < 4; ++g) gs[g] = sc[2 * g] + sc[2 * g + 1];
        int g1 = 0; for (int g = 1; g < 4; ++g) if (gs[g] > gs[g1]) g1 = g;
        int g2 = -1;
        for (int g = 0; g < 4; ++g) { if (g == g1) continue; if (g2 < 0 || gs[g] > gs[g2]) g2 = g; }
        float masked[8];
        for (int e = 0; e < 8; ++e) {
            const int g = e >> 1;
            masked[e] = (g == g1 || g == g2) ? sc[e] : 0.0f;
        }
        int i1 = 0; for (int e = 1; e < 8; ++e) if (masked[e] > masked[i1]) i1 = e;
        int i2 = -1;
        for (int e = 0; e < 8; ++e) { if (e == i1) continue; if (i2 < 0 || masked[e] > masked[i2]) i2 = e; }
        const float w1 = sr[i1], w2 = sr[i2];
        const float inv = 2.5f / (w1 + w2 + 1e-20f);
        for (int e = 0; e < 8; ++e) dw[t * 8 + e] = 0.f;
        dw[t * 8 + i1] = w1 * inv;
        dw[t * 8 + i2] = w2 * inv;
    }
}

__global__ void silumul_kernel(const float* __restrict__ g,
                               const float* __restrict__ u,
                               float* __restrict__ a, int n)
{
    const int i = blockIdx.x * blockDim.x + threadIdx.x;
    if (i < n) { const float x = g[i]; a[i] = (x / (1.f + __expf(-x))) * u[i]; }
}

// =====================================================================
static inline void launch_gemm(const float* A, int lda, const float* W, int ldw,
                               float* C, int ldc, int M, int N, int K, int mode,
                               const float* rs, int rss, hipStream_t st)
{
    dim3 grid(N / GBN, M / GBM);
    gemm_wmma_f32<<<grid, 256, 0, st>>>(A, lda, W, ldw, C, ldc, M, N, K, mode, rs, rss);
}

extern "C" void kernel_launch(void* const* d_in, const int* in_sizes, int n_in,
                              void* d_out, int out_size, void* d_ws, size_t ws_size,
                              hipStream_t stream)
{
    const int B = 2, S = 1024, T = B * S, H = H_C;

    const float* hidden   = (const float*)d_in[0];
    const float* cosb     = (const float*)d_in[1];
    const float* sinb     = (const float*)d_in[2];
    const float* ln1_w    = (const float*)d_in[3];
    const float* q_a_w    = (const float*)d_in[4];
    const float* q_a_ln_w = (const float*)d_in[5];
    const float* q_b_w    = (const float*)d_in[6];
    const float* kv_a_w   = (const float*)d_in[7];
    const float* kv_a_ln  = (const float*)d_in[8];
    const float* kv_b_w   = (const float*)d_in[9];
    const float* o_w      = (const float*)d_in[10];
    const float* ln2_w    = (const float*)d_in[11];
    const float* router_w = (const float*)d_in[12];
    const float* router_b = (const float*)d_in[13];
    const float* gate_w   = (const float*)d_in[14];
    const float* up_w     = (const float*)d_in[15];
    const float* down_w   = (const float*)d_in[16];
    const float* sh_gate  = (const float*)d_in[17];
    const float* sh_up    = (const float*)d_in[18];
    const float* sh_down  = (const float*)d_in[19];
    float* out = (float*)d_out;

    // ---- workspace carve-up (floats) ----
    float* w = (float*)d_ws;
    size_t off = 0;
    float* xln  = w + off; off += (size_t)T * H;
    float* qa   = w + off; off += (size_t)T * Q_LORA_C;
    float* qb   = w + off; off += (size_t)T * (NH_C * DQK_C);
    float* ckv  = w + off; off += (size_t)T * 576;
    float* kvb  = w + off; off += (size_t)T * (NH_C * 256);
    float* attn = w + off; off += (size_t)T * (NH_C * DV_C);
    float* x2   = w + off; off += (size_t)T * H;
    float* dw   = w + off; off += (size_t)T * N_EXP_C;
    // aliases (lifetimes disjoint)
    float* hbuf  = xln;                 // ln2 output, after attention path done with xln
    float* gateb = attn;                // attn dead after o-proj
    float* upb   = attn + (size_t)T * INTER_C;
    float* actb  = qa;                  // qa dead after q_b GEMM
    (void)ws_size; (void)in_sizes; (void)n_in; (void)out_size;

    // ---- attention branch ----
    rmsnorm_kernel<<<T, 256, 0, stream>>>(hidden, H, xln, H, ln1_w, H);
    launch_gemm(xln, H, q_a_w, H, qa, Q_LORA_C, T, Q_LORA_C, H, 0, nullptr, 0, stream);
    rmsnorm_kernel<<<T, 256, 0, stream>>>(qa, Q_LORA_C, qa, Q_LORA_C, q_a_ln_w, Q_LORA_C);
    launch_gemm(qa, Q_LORA_C, q_b_w, Q_LORA_C, qb, NH_C * DQK_C,
                T, NH_C * DQK_C, Q_LORA_C, 0, nullptr, 0, stream);
    launch_gemm(xln, H, kv_a_w, H, ckv, 576, T, 576, H, 0, nullptr, 0, stream);
    rmsnorm_kernel<<<T, 256, 0, stream>>>(ckv, 576, ckv, 576, kv_a_ln, KV_LORA_C);
    rope_kernel<<<T, 256, 0, stream>>>(qb, ckv, cosb, sinb);
    launch_gemm(ckv, 576, kv_b_w, KV_LORA_C, kvb, NH_C * 256,
                T, NH_C * 256, KV_LORA_C, 0, nullptr, 0, stream);
    attn_kernel<<<dim3(S / 64, NH_C, B), 128, 0, stream>>>(qb, kvb, ckv, attn, S);

    hipMemcpyAsync(x2, hidden, (size_t)T * H * sizeof(float),
                   hipMemcpyDeviceToDevice, stream);
    launch_gemm(attn, NH_C * DV_C, o_w, NH_C * DV_C, x2, H,
                T, H, NH_C * DV_C, 1, nullptr, 0, stream);

    // ---- MoE branch ----
    rmsnorm_kernel<<<T, 256, 0, stream>>>(x2, H, hbuf, H, ln2_w, H);
    router_kernel<<<T, 256, 0, stream>>>(hbuf, router_w, router_b, dw, H);
    hipMemcpyAsync(out, x2, (size_t)T * H * sizeof(float),
                   hipMemcpyDeviceToDevice, stream);

    const int nElem = T * INTER_C;
    for (int e = 0; e < N_EXP_C; ++e) {
        const float* gw  = gate_w + (size_t)e * INTER_C * H;
        const float* uw  = up_w   + (size_t)e * INTER_C * H;
        const float* dwn = down_w + (size_t)e * H * INTER_C;
        launch_gemm(hbuf, H, gw, H, gateb, INTER_C, T, INTER_C, H, 0, nullptr, 0, stream);
        launch_gemm(hbuf, H, uw, H, upb, INTER_C, T, INTER_C, H, 0, nullptr, 0, stream);
        silumul_kernel<<<(nElem + 255) / 256, 256, 0, stream>>>(gateb, upb, actb, nElem);
        launch_gemm(actb, INTER_C, dwn, INTER_C, out, H,
                    T, H, INTER_C, 2, dw + e, N_EXP_C, stream);
    }

    // ---- shared expert ----
    launch_gemm(hbuf, H, sh_gate, H, gateb, INTER_C, T, INTER_C, H, 0, nullptr, 0, stream);
    launch_gemm(hbuf, H, sh_up,   H, upb,   INTER_C, T, INTER_C, H, 0, nullptr, 0, stream);
    silumul_kernel<<<(nElem + 255) / 256, 256, 0, stream>>>(gateb, upb, actb, nElem);
    launch_gemm(actb, INTER_C, sh_down, INTER_C, out, H, T, H, INTER_C, 1, nullptr, 0, stream);
}